// Abstraction_80977313398939
// MI455X (gfx1250) — compile-verified
//
#include <hip/hip_runtime.h>
#include <hip/hip_bf16.h>

// Problem constants (match reference setup_inputs).
constexpr int Bn   = 4;            // batch
constexpr int Nn   = 4096;         // sequence
constexpr int Dn   = 512;          // feature dim
constexpr int ABS  = 64;           // ABSTRACTION_SIZE
constexpr int Qn   = Nn / ABS;     // 64 blocks of 64 rows

typedef float v2f __attribute__((ext_vector_type(2)));
typedef float v8f __attribute__((ext_vector_type(8)));

// ---------------------------------------------------------------------------
// fp32 WMMA 16x16x4 (CDNA5): D = A(16x4) x B(4x16) + C, fp32 throughout.
// ---------------------------------------------------------------------------
__device__ __forceinline__ v8f wmma4(v2f a, v2f b, v8f c) {
    return __builtin_amdgcn_wmma_f32_16x16x4_f32(false, a, false, b,
                                                 (short)0, c, false, false);
}

// A-fragment for row-major M x K matrix, tile origin p = &M[row0*ld + k0].
// ISA layout (32-bit A 16x4): lane l, VGPR j holds A[l%16][k0 + j + 2*(l>>4)]
// -> one contiguous float2 per lane. Also serves as the B-fragment loader for
// A @ B^T when B is stored row-major / transposed-in-LDS [N, K].
__device__ __forceinline__ v2f frag_rowmajor(const float* p, int ld, int lane) {
    const float* q = p + (size_t)(lane & 15) * ld + 2 * (lane >> 4);
    v2f r; r.x = q[0]; r.y = q[1]; return r;
}

// B-fragment for row-major K x N matrix, tile origin p = &B[k0*ld + n0].
__device__ __forceinline__ v2f frag_kn(const float* p, int ld, int lane) {
    const float* q = p + (size_t)(2 * (lane >> 4)) * ld + (lane & 15);
    v2f r; r.x = q[0]; r.y = q[ld]; return r;
}

// ---------------------------------------------------------------------------
// K1: c = l2_normalize(data, axis=-1)   (tf semantics: x * rsqrt(max(ss,eps)))
// ---------------------------------------------------------------------------
__global__ __launch_bounds__(128)
void normalize_kernel(const float* __restrict__ data, float* __restrict__ c) {
    __shared__ float red[128];
    __shared__ float s_inv;
    const int row = blockIdx.x;
    const int t   = threadIdx.x;
    const float4 v = ((const float4*)(data + (size_t)row * Dn))[t];
    red[t] = v.x * v.x + v.y * v.y + v.z * v.z + v.w * v.w;
    __syncthreads();
    for (int s = 64; s > 0; s >>= 1) {
        if (t < s) red[t] += red[t + s];
        __syncthreads();
    }
    if (t == 0) s_inv = rsqrtf(fmaxf(red[0], 1e-12f));
    __syncthreads();
    const float inv = s_inv;
    float4 o; o.x = v.x * inv; o.y = v.y * inv; o.z = v.z * inv; o.w = v.w * inv;
    ((float4*)(c + (size_t)row * Dn))[t] = o;
}

// ---------------------------------------------------------------------------
// K2: block-prefix sums  P[b,q,a,d] = sum_{g<q} c[b, g*64+a, d]
// ---------------------------------------------------------------------------
__global__ __launch_bounds__(256)
void prefix_kernel(const float* __restrict__ c, float* __restrict__ P) {
    const int t = blockIdx.x * blockDim.x + threadIdx.x;   // 0 .. B*ABS*D-1
    const int d = t & (Dn - 1);
    const int a = (t >> 9) & (ABS - 1);
    const int b = t >> 15;
    const float* cp = c + ((size_t)b * Nn + a) * Dn + d;   // rows n = q*64+a
    float*       pp = P + ((size_t)b * Qn * ABS + a) * Dn + d;
    float run = 0.0f;
    for (int q = 0; q < Qn; ++q) {
        pp[(size_t)q * ABS * Dn] = run;
        run += cp[(size_t)q * ABS * Dn];
    }
}

// ---------------------------------------------------------------------------
// K3: pooled[b, q*64+r, a] = (U[r,a] + (a<=r) * V[r,a]) / 64
//     U = C_q @ P_q^T,  V = C_q @ C_q^T   (two 64x64x512 GEMMs per block q)
// One block per (b,q); 8 waves; each wave computes 2 (U,V) 16x16 tile pairs.
// ---------------------------------------------------------------------------
__global__ __launch_bounds__(256)
void pooled_kernel(const float* __restrict__ c, const float* __restrict__ P,
                   float* __restrict__ pooled) {
    const int bq   = blockIdx.x;               // 0..255
    const int b    = bq >> 6;
    const int q    = bq & 63;
    const int wave = threadIdx.x >> 5;
    const int lane = threadIdx.x & 31;
    const float* cblk = c + ((size_t)b * Nn + (size_t)q * ABS) * Dn;  // 64 x 512
    const float* pblk = P + (size_t)bq * ABS * Dn;                    // 64 x 512
    float* outblk = pooled + ((size_t)b * Nn + (size_t)q * ABS) * ABS;

    for (int pair = wave; pair < 16; pair += 8) {       // uniform per wave
        const int ti = pair >> 2, tj = pair & 3;
        v8f au = {}; v8f av = {};
        const float* arow  = cblk + (size_t)ti * 16 * Dn;
        const float* burow = pblk + (size_t)tj * 16 * Dn;   // rows a (P_q^T)
        const float* bvrow = cblk + (size_t)tj * 16 * Dn;   // rows a (C_q^T)
        #pragma unroll 4
        for (int k = 0; k < Dn; k += 4) {
            const v2f af = frag_rowmajor(arow  + k, Dn, lane);
            const v2f bu = frag_rowmajor(burow + k, Dn, lane);
            const v2f bv = frag_rowmajor(bvrow + k, Dn, lane);
            au = wmma4(af, bu, au);
            av = wmma4(af, bv, av);
        }
        const int rbase = ti * 16 + 8 * (lane >> 4);
        const int a     = tj * 16 + (lane & 15);
        #pragma unroll
        for (int jj = 0; jj < 8; ++jj) {
            const int r = rbase + jj;
            const float val = au[jj] + ((a <= r) ? av[jj] : 0.0f);
            outblk[(size_t)r * ABS + a] = val * (1.0f / 64.0f);
        }
    }
}

// ---------------------------------------------------------------------------
// K4: W_fused[64,512] = W_abs @ W_merge[512:1024, :]   (tiny GEMM, K=512)
// ---------------------------------------------------------------------------
__global__ __launch_bounds__(256)
void fusew_kernel(const float* __restrict__ Wabs, const float* __restrict__ Wm,
                  float* __restrict__ Wf) {
    const int tile = blockIdx.x * 8 + (threadIdx.x >> 5);  // 0..127
    const int lane = threadIdx.x & 31;
    const int tm = tile >> 5;                              // 0..3
    const int tn = tile & 31;                              // 0..31
    v8f acc = {};
    const float* arow = Wabs + (size_t)tm * 16 * Dn;
    const float* bcol = Wm + (size_t)Dn * Dn + tn * 16;    // W2 = rows 512..1023
    #pragma unroll 4
    for (int k = 0; k < Dn; k += 4)
        acc = wmma4(frag_rowmajor(arow + k, Dn, lane),
                    frag_kn(bcol + (size_t)k * Dn, Dn, lane), acc);
    float* o = Wf + (size_t)(tm * 16 + 8 * (lane >> 4)) * Dn + tn * 16 + (lane & 15);
    #pragma unroll
    for (int jj = 0; jj < 8; ++jj) o[(size_t)jj * Dn] = acc[jj];
}

// ---------------------------------------------------------------------------
// K5: out = data @ W1 + pooled @ W_fused        ([16384,512] result)
// Block = 128 rows x 64 cols; 8 waves, each wave a 16x64 strip (4 tiles).
// Weight slabs staged transposed in LDS (pad 513/65 -> conflict-free),
// so B fragments are single contiguous ds_load_b64 per tile per k-step.
// L2 traffic: data 268 MB (4x reuse in regs), weights 134 MB (LDS reuse).
// ---------------------------------------------------------------------------
constexpr int WT_LD  = 513;                        // padded ld for W1 slab
constexpr int WFT_LD = 65;                         // padded ld for Wf slab
constexpr int MERGE_LDS_FLOATS = 64 * WT_LD + 64 * WFT_LD;

__global__ __launch_bounds__(256)
void merge_kernel(const float* __restrict__ data, const float* __restrict__ Wm,
                  const float* __restrict__ pooled, const float* __restrict__ Wf,
                  float* __restrict__ out) {
    extern __shared__ float smem[];
    float* Wt  = smem;                 // [64][513]: Wt[n][k] = W1[k][tng*64+n]
    float* Wft = smem + 64 * WT_LD;    // [64][65] : Wft[n][k] = Wf[k][tng*64+n]

    const int tmg = blockIdx.x >> 3;   // 0..127 row group (8 tiles = 128 rows)
    const int tng = blockIdx.x & 7;    // 0..7   col group (64 columns)
    const int tid = threadIdx.x;

    // Stage transposed slabs: coalesced global reads (consecutive n),
    // bank-conflict-free LDS writes thanks to odd ld.
    for (int idx = tid; idx < Dn * 64; idx += 256) {
        const int n = idx & 63, k = idx >> 6;
        Wt[n * WT_LD + k] = Wm[(size_t)k * Dn + tng * 64 + n];   // W1 rows 0..511
    }
    for (int idx = tid; idx < ABS * 64; idx += 256) {
        const int n = idx & 63, k = idx >> 6;
        Wft[n * WFT_LD + k] = Wf[(size_t)k * Dn + tng * 64 + n];
    }
    __syncthreads();

    const int wave = tid >> 5, lane = tid & 31;
    const int tm = tmg * 8 + wave;                 // 0..1023
    v8f acc0 = {}, acc1 = {}, acc2 = {}, acc3 = {};

    // data @ W1, K = 512: 1 global b64 + 4 ds b64 feed 4 WMMAs per k-step.
    const float* arow = data + (size_t)tm * 16 * Dn;
    #pragma unroll 2
    for (int k = 0; k < Dn; k += 4) {
        const v2f a = frag_rowmajor(arow + k, Dn, lane);
        acc0 = wmma4(a, frag_rowmajor(Wt + 0 * 16 * WT_LD + k, WT_LD, lane), acc0);
        acc1 = wmma4(a, frag_rowmajor(Wt + 1 * 16 * WT_LD + k, WT_LD, lane), acc1);
        acc2 = wmma4(a, frag_rowmajor(Wt + 2 * 16 * WT_LD + k, WT_LD, lane), acc2);
        acc3 = wmma4(a, frag_rowmajor(Wt + 3 * 16 * WT_LD + k, WT_LD, lane), acc3);
    }
    // pooled @ W_fused, K = 64.
    const float* prow = pooled + (size_t)tm * 16 * ABS;
    #pragma unroll
    for (int k = 0; k < ABS; k += 4) {
        const v2f a = frag_rowmajor(prow + k, ABS, lane);
        acc0 = wmma4(a, frag_rowmajor(Wft + 0 * 16 * WFT_LD + k, WFT_LD, lane), acc0);
        acc1 = wmma4(a, frag_rowmajor(Wft + 1 * 16 * WFT_LD + k, WFT_LD, lane), acc1);
        acc2 = wmma4(a, frag_rowmajor(Wft + 2 * 16 * WFT_LD + k, WFT_LD, lane), acc2);
        acc3 = wmma4(a, frag_rowmajor(Wft + 3 * 16 * WFT_LD + k, WFT_LD, lane), acc3);
    }

    // Writeout: rows tm*16 + jj + 8*(lane>>4), cols tng*64 + j*16 + lane%16.
    float* obase = out + (size_t)(tm * 16 + 8 * (lane >> 4)) * Dn
                 + tng * 64 + (lane & 15);
    #pragma unroll
    for (int jj = 0; jj < 8; ++jj) {
        float* o = obase + (size_t)jj * Dn;
        o[0]  = acc0[jj];
        o[16] = acc1[jj];
        o[32] = acc2[jj];
        o[48] = acc3[jj];
    }
}

// ---------------------------------------------------------------------------
extern "C" void kernel_launch(void* const* d_in, const int* in_sizes, int n_in,
                              void* d_out, int out_size, void* d_ws, size_t ws_size,
                              hipStream_t stream) {
    const float* data = (const float*)d_in[0];   // [4,4096,512]
    const float* Wabs = (const float*)d_in[1];   // [64,512]
    const float* Wm   = (const float*)d_in[2];   // [1024,512]
    float* out = (float*)d_out;                  // [4,4096,512]

    float* ws     = (float*)d_ws;
    float* c      = ws;                                          // B*N*D   (33.5 MB)
    float* P      = c      + (size_t)Bn * Nn * Dn;               // B*Q*A*D (33.5 MB)
    float* pooled = P      + (size_t)Bn * Qn * ABS * Dn;         // B*N*A   ( 4.2 MB)
    float* Wf     = pooled + (size_t)Bn * Nn * ABS;              // A*D     ( 128 KB)

    normalize_kernel<<<Bn * Nn, 128, 0, stream>>>(data, c);
    prefix_kernel<<<(Bn * ABS * Dn) / 256, 256, 0, stream>>>(c, P);
    pooled_kernel<<<Bn * Qn, 256, 0, stream>>>(c, P, pooled);
    fusew_kernel<<<16, 256, 0, stream>>>(Wabs, Wm, Wf);
    merge_kernel<<<1024, 256, MERGE_LDS_FLOATS * sizeof(float), stream>>>(
        data, Wm, pooled, Wf, out);
}